// YoloLossLayer_88184268521538
// MI455X (gfx1250) — compile-verified
//
#include <hip/hip_runtime.h>
#include <math.h>

#define NCLS 15
#define NA   18
#define NG   52
#define CH   (NCLS + 6)          // 21 channels per anchor
#define CPI  (NG * NG)           // 2704 cells per (b,a) plane
#define EPSF 1e-7f

typedef float v2f __attribute__((ext_vector_type(2)));
typedef float v8f __attribute__((ext_vector_type(8)));

// ma = [[w/8, h/8, angle] for (w,h) in ANCHORS for angle in ANGLES]
__device__ __constant__ float d_maw[NA] = {
    1.25f, 1.25f, 1.25f, 1.25f, 1.25f, 1.25f,
    2.0f,  2.0f,  2.0f,  2.0f,  2.0f,  2.0f,
    4.125f,4.125f,4.125f,4.125f,4.125f,4.125f };
__device__ __constant__ float d_mah[NA] = {
    1.625f,1.625f,1.625f,1.625f,1.625f,1.625f,
    3.75f, 3.75f, 3.75f, 3.75f, 3.75f, 3.75f,
    2.875f,2.875f,2.875f,2.875f,2.875f,2.875f };
__device__ __constant__ float d_maa[NA] = {
    -1.0471975512f, -0.5235987756f, 0.0f, 0.5235987756f, 1.0471975512f, 1.5707963268f,
    -1.0471975512f, -0.5235987756f, 0.0f, 0.5235987756f, 1.0471975512f, 1.5707963268f,
    -1.0471975512f, -0.5235987756f, 0.0f, 0.5235987756f, 1.0471975512f, 1.5707963268f };

__device__ __forceinline__ float sigmoidf(float v) { return 1.0f / (1.0f + expf(-v)); }
__device__ __forceinline__ float clip01(float p) {
    return fminf(fmaxf(p, EPSF), 1.0f - EPSF);
}

// Sutherland-Hodgman skew IoU, matching reference _clip/_area semantics
// (d>=0 inside, den clamp at 1e-12, K=8 cap with drop).
__device__ float skewiou5(const float b1[5], const float b2[5]) {
    float c1x[4], c1y[4], c2x[4], c2y[4];
    const float dxs[4] = {0.5f, -0.5f, -0.5f, 0.5f};
    const float dys[4] = {0.5f, 0.5f, -0.5f, -0.5f};
    {
        float ca = cosf(b1[4]), sa = sinf(b1[4]);
        #pragma unroll
        for (int i = 0; i < 4; i++) {
            float X = dxs[i] * b1[2], Y = dys[i] * b1[3];
            c1x[i] = b1[0] + ca * X - sa * Y;
            c1y[i] = b1[1] + sa * X + ca * Y;
        }
    }
    {
        float ca = cosf(b2[4]), sa = sinf(b2[4]);
        #pragma unroll
        for (int i = 0; i < 4; i++) {
            float X = dxs[i] * b2[2], Y = dys[i] * b2[3];
            c2x[i] = b2[0] + ca * X - sa * Y;
            c2y[i] = b2[1] + sa * X + ca * Y;
        }
    }
    float px[8], py[8];
    int n = 4;
    #pragma unroll
    for (int i = 0; i < 4; i++) { px[i] = c2x[i]; py[i] = c2y[i]; }
    #pragma unroll
    for (int i = 4; i < 8; i++) { px[i] = 0.0f; py[i] = 0.0f; }

    for (int e = 0; e < 4; e++) {
        float ax = c1x[e], ay = c1y[e];
        float bx = c1x[(e + 1) & 3], by = c1y[(e + 1) & 3];
        float ex = bx - ax, ey = by - ay;
        float d[8];
        #pragma unroll
        for (int i = 0; i < 8; i++) d[i] = ex * (py[i] - ay) - ey * (px[i] - ax);
        float ox[8], oy[8];
        int m = 0;
        for (int i = 0; i < 8; i++) {
            if (i >= n) break;
            int nx = i + 1; if (nx >= n) nx = 0;
            bool ins  = d[i]  >= 0.0f;
            bool insn = d[nx] >= 0.0f;
            if (ins && m < 8) { ox[m] = px[i]; oy[m] = py[i]; m++; }
            if (ins != insn) {
                float den = d[i] - d[nx];
                if (fabsf(den) < 1e-12f) den = 1e-12f;
                float tt = d[i] / den;
                if (m < 8) {
                    ox[m] = px[i] + tt * (px[nx] - px[i]);
                    oy[m] = py[i] + tt * (py[nx] - py[i]);
                    m++;
                }
            }
        }
        n = m;
        #pragma unroll
        for (int i = 0; i < 8; i++) {
            px[i] = (i < m) ? ox[i] : 0.0f;
            py[i] = (i < m) ? oy[i] : 0.0f;
        }
    }
    float area = 0.0f;
    if (n > 0) {
        for (int i = 0; i < 8; i++) {
            if (i >= n) break;
            int nx = i + 1; if (nx >= n) nx = 0;
            area += px[i] * py[nx] - px[nx] * py[i];
        }
        area = 0.5f * fabsf(area);
    }
    return area / (b1[2] * b1[3] + b2[2] * b2[3] - area + 1e-9f);
}

// ---------------- kernel 1: zero flags + owner tables ----------------
__global__ void k_init(unsigned* __restrict__ flags, unsigned* __restrict__ owner, int NC) {
    int i = blockIdx.x * blockDim.x + threadIdx.x;
    if (i < NC) { flags[i] = 0u; owner[i] = 0u; }
}

// ---------------- kernel 2: fully parallel per-target work ----------------
// One thread per target: anchor IoUs, ignore scatter, best_n, box decode,
// CIoU + skew IoU, deterministic owner election (atomicMax, last target wins).
__global__ void k_target(const float* __restrict__ x, const float* __restrict__ tgt,
                         int nT, unsigned* __restrict__ flags, unsigned* __restrict__ owner,
                         int* __restrict__ cell_t, float* __restrict__ ta_t,
                         float* __restrict__ ar_t, float* __restrict__ ci_t) {
    int t = blockIdx.x * blockDim.x + threadIdx.x;
    if (t >= nT) return;

    float tx = tgt[t * 7 + 0] * (float)NG;
    float ty = tgt[t * 7 + 1] * (float)NG;
    float tw = tgt[t * 7 + 2] * (float)NG;
    float th = tgt[t * 7 + 3] * (float)NG;
    float ga = tgt[t * 7 + 4];
    int lab  = (int)tgt[t * 7 + 5];
    int b    = (int)tgt[t * 7 + 6];
    int gi = (int)tx, gj = (int)ty;

    // anchor-vs-target skew IoUs, ignore scatter, argmax (first max wins)
    int best = 0;
    float bi = -1.0f;
    float gt0[5] = {0.0f, 0.0f, tw, th, ga};
    for (int a = 0; a < NA; a++) {
        float an0[5] = {0.0f, 0.0f, d_maw[a], d_mah[a], d_maa[a]};
        float iou = skewiou5(an0, gt0);
        if (iou > bi) { bi = iou; best = a; }
        float off = fabsf(d_maa[a] - ga);
        bool ign = (iou > 0.6f) || ((iou > 0.3f) && (off < 0.2618f));
        if (ign) {
            int icell = ((b * NA + a) * NG + gj) * NG + gi;
            atomicOr(&flags[icell], 2u);
        }
    }

    int cell = ((b * NA + best) * NG + gj) * NG + gi;
    long base = ((long)(b * NA + best)) * CH * CPI + (long)gj * NG + gi;
    float c0 = x[base + 0L * CPI];
    float c1 = x[base + 1L * CPI];
    float c2 = x[base + 2L * CPI];
    float c3 = x[base + 3L * CPI];
    float c4 = x[base + 4L * CPI];
    float pxc = sigmoidf(c0) + (float)gi;
    float pyc = sigmoidf(c1) + (float)gj;
    float pw  = expf(c2) * d_maw[best];
    float ph  = expf(c3) * d_mah[best];
    float pa  = c4 + d_maa[best];

    // CIoU (forward value)
    float iw = fminf(pxc + pw * 0.5f, tx + tw * 0.5f) -
               fmaxf(pxc - pw * 0.5f, tx - tw * 0.5f);
    iw = fmaxf(iw, 0.0f);
    float ih = fminf(pyc + ph * 0.5f, ty + th * 0.5f) -
               fmaxf(pyc - ph * 0.5f, ty - th * 0.5f);
    ih = fmaxf(ih, 0.0f);
    float inter = iw * ih;
    float iou = inter / (pw * ph + tw * th - inter + EPSF);
    float cw  = fmaxf(pxc + pw * 0.5f, tx + tw * 0.5f) -
                fminf(pxc - pw * 0.5f, tx - tw * 0.5f);
    float chh = fmaxf(pyc + ph * 0.5f, ty + th * 0.5f) -
                fminf(pyc - ph * 0.5f, ty - th * 0.5f);
    float c2d = cw * cw + chh * chh + EPSF;
    float rho2 = (pxc - tx) * (pxc - tx) + (pyc - ty) * (pyc - ty);
    const float PI = 3.14159265358979323846f;
    float dat = atanf(tw / (th + EPSF)) - atanf(pw / (ph + EPSF));
    float v = (4.0f / (PI * PI)) * dat * dat;
    float alpha = v / (1.0f - iou + v + EPSF);
    float ciou = iou - rho2 / c2d - alpha * v;

    float pb5[5] = {pxc, pyc, pw, ph, pa};
    float tb5[5] = {tx, ty, tw, th, ga};
    float ar = skewiou5(pb5, tb5);

    cell_t[t] = cell;
    ta_t[t] = ga - d_maa[best];
    ar_t[t] = expf(1.0f - ar) - 1.0f;
    ci_t[t] = 1.0f - ciou;
    atomicOr(&flags[cell], 1u | (1u << (2 + lab)));
    atomicMax(&owner[cell], (unsigned)(t + 1));   // order-independent -> deterministic
}

// ---------------- kernel 3: obj-cell reg + cls loss (winner threads only) ----------------
__launch_bounds__(64)
__global__ void k_objloss(const float* __restrict__ x, const unsigned* __restrict__ flags,
                          const unsigned* __restrict__ owner, const int* __restrict__ cell_t,
                          const float* __restrict__ ta_t, const float* __restrict__ ar_t,
                          const float* __restrict__ ci_t, int nT,
                          float* __restrict__ objpart) {
    int t = blockIdx.x * blockDim.x + threadIdx.x;
    float srg = 0.0f, scl = 0.0f;
    if (t < nT) {
        int cell = cell_t[t];
        if (owner[cell] == (unsigned)(t + 1)) {   // exactly one winner per obj cell
            unsigned f = flags[cell];
            int ba = cell / CPI;
            int ji = cell - ba * CPI;
            long base = (long)ba * CH * CPI + ji;
            float pa = x[base + 4L * CPI];
            float dd = pa - ta_t[t];
            float ad = fabsf(dd);
            float sl1 = (ad < 1.0f) ? (0.5f * dd * dd) : (ad - 0.5f);
            float reg = sl1 + ci_t[t];
            float regc = ar_t[t] / reg;           // stop_gradient forward value
            srg = reg * regc;
            #pragma unroll
            for (int c = 0; c < NCLS; c++) {
                float p = clip01(sigmoidf(x[base + (long)(6 + c) * CPI]));
                float tt = ((f >> (2 + c)) & 1u) ? 1.0f : 0.0f;
                scl += -(tt * logf(p) + (1.0f - tt) * logf(1.0f - p));
            }
        }
    }
    // block reduce (64 threads = 2 waves), fixed order -> deterministic
    #pragma unroll
    for (int o = 16; o > 0; o >>= 1) {
        srg += __shfl_xor(srg, o, 32);
        scl += __shfl_xor(scl, o, 32);
    }
    __shared__ float smem[2][2];
    int wave = threadIdx.x >> 5, lane = threadIdx.x & 31;
    if (lane == 0) { smem[wave][0] = srg; smem[wave][1] = scl; }
    __syncthreads();
    if (threadIdx.x == 0) {
        objpart[blockIdx.x * 2 + 0] = smem[0][0] + smem[1][0];
        objpart[blockIdx.x * 2 + 1] = smem[0][1] + smem[1][1];
    }
}

// ---------------- kernel 4: dense conf loss + WMMA block reduction ----------------
__launch_bounds__(256)
__global__ void k_loss(const float* __restrict__ x, const unsigned* __restrict__ flags,
                       float* __restrict__ partials, int NC) {
    int cell = blockIdx.x * blockDim.x + threadIdx.x;
    float sco = 0.0f, scn = 0.0f, cno = 0.0f, cnn = 0.0f;
    if (cell < NC) {
        unsigned f = flags[cell];
        bool obj = (f & 1u) != 0u;
        bool ign = (f & 2u) != 0u;
        bool noobj = (!obj) && (!ign);
        if (obj || noobj) {
            int ba = cell / CPI;            // b*NA + a
            int ji = cell - ba * CPI;       // j*NG + i
            long base = (long)ba * CH * CPI + ji;
            float p = clip01(sigmoidf(x[base + 5L * CPI]));
            if (obj) { sco = -logf(p); cno = 1.0f; }
            else     { scn = -logf(1.0f - p); cnn = 1.0f; }
        }
    }
    // per-wave f32 butterfly reduction (exact order, deterministic)
    float acc[4] = {sco, scn, cno, cnn};
    #pragma unroll
    for (int j = 0; j < 4; j++)
        #pragma unroll
        for (int o = 16; o > 0; o >>= 1)
            acc[j] += __shfl_xor(acc[j], o, 32);

    __shared__ float smem[8][4];
    int wave = threadIdx.x >> 5;
    int lane = threadIdx.x & 31;
    if (lane == 0) {
        #pragma unroll
        for (int j = 0; j < 4; j++) smem[wave][j] = acc[j];
    }
    __syncthreads();

    if (threadIdx.x < 32) {   // full wave 0, EXEC all ones -> WMMA legal
        int l = threadIdx.x;
        int row  = (l < 16) ? l : (l - 16);
        int accI = (row < 8) ? row : (row - 8);
        // A[row][K]: rows 0-7 hold acc=row over waves K=0..3; rows 8-15 hold waves 4..7
        int kbase = ((row < 8) ? 0 : 4) + ((l < 16) ? 0 : 2);
        float ax = (accI < 4) ? smem[kbase][accI] : 0.0f;
        float ay = (accI < 4) ? smem[kbase + 1][accI] : 0.0f;
#if __has_builtin(__builtin_amdgcn_wmma_f32_16x16x4_f32)
        v2f a; a[0] = ax; a[1] = ay;
        v2f b; b[0] = 1.0f; b[1] = 1.0f;          // B = all ones -> D[m,n] = sum_k A[m,k]
        v8f cz = {};
        v8f dmat = __builtin_amdgcn_wmma_f32_16x16x4_f32(
            /*neg_a=*/false, a, /*neg_b=*/false, b,
            /*c_mod=*/(short)0, cz, /*reuse_a=*/false, /*reuse_b=*/false);
        float tot[4];
        #pragma unroll
        for (int j = 0; j < 4; j++) {
            float hi = __shfl_xor(dmat[j], 16, 32);   // fold rows 8..15
            tot[j] = dmat[j] + hi;
        }
        if (l == 0) {
            #pragma unroll
            for (int j = 0; j < 4; j++) partials[(long)blockIdx.x * 4 + j] = tot[j];
        }
#else
        (void)ax; (void)ay; (void)accI; (void)kbase;
        if (l == 0) {
            #pragma unroll
            for (int j = 0; j < 4; j++) {
                float s = 0.0f;
                for (int w = 0; w < 8; w++) s += smem[w][j];
                partials[(long)blockIdx.x * 4 + j] = s;
            }
        }
#endif
    }
}

// ---------------- kernel 5: final reduction + loss scalars ----------------
__launch_bounds__(256)
__global__ void k_final(const float* __restrict__ partials, int nblocks,
                        const float* __restrict__ objpart, int noblk,
                        int bs, float* __restrict__ out) {
    float acc[6] = {0, 0, 0, 0, 0, 0};   // conf_obj, conf_noobj, n_obj, n_noobj, srg, scl
    for (int i = threadIdx.x; i < nblocks; i += blockDim.x) {
        #pragma unroll
        for (int j = 0; j < 4; j++) acc[j] += partials[(long)i * 4 + j];
    }
    for (int i = threadIdx.x; i < noblk; i += blockDim.x) {
        acc[4] += objpart[i * 2 + 0];
        acc[5] += objpart[i * 2 + 1];
    }
    #pragma unroll
    for (int j = 0; j < 6; j++)
        #pragma unroll
        for (int o = 16; o > 0; o >>= 1)
            acc[j] += __shfl_xor(acc[j], o, 32);
    __shared__ float smem[8][6];
    int wave = threadIdx.x >> 5, lane = threadIdx.x & 31;
    if (lane == 0) {
        #pragma unroll
        for (int j = 0; j < 6; j++) smem[wave][j] = acc[j];
    }
    __syncthreads();
    if (threadIdx.x == 0) {
        float tot[6];
        #pragma unroll
        for (int j = 0; j < 6; j++) {
            float s = 0.0f;
            for (int w = 0; w < 8; w++) s += smem[w][j];
            tot[j] = s;
        }
        float n_obj   = fmaxf(tot[2], 1.0f);
        float n_noobj = fmaxf(tot[3], 1.0f);
        float reg_loss  = tot[4] / n_obj;
        float conf_loss = tot[0] / n_obj + tot[1] / n_noobj;
        float cls_loss  = tot[5] / (n_obj * (float)NCLS);
        float total = reg_loss + conf_loss + cls_loss;
        float fb = (float)bs;
        out[0] = total * fb;
        out[1] = reg_loss * fb;
        out[2] = conf_loss * fb;
        out[3] = cls_loss * fb;
        out[4] = (float)(NG * NG * NA);
    }
}

extern "C" void kernel_launch(void* const* d_in, const int* in_sizes, int n_in,
                              void* d_out, int out_size, void* d_ws, size_t ws_size,
                              hipStream_t stream) {
    const float* x   = (const float*)d_in[0];
    const float* tgt = (const float*)d_in[1];
    int bs = in_sizes[0] / (NA * CH * NG * NG);   // 16
    int nT = in_sizes[1] / 7;                     // 200
    int NC = bs * NA * NG * NG;                   // 778,752

    char* ws = (char*)d_ws;
    unsigned* flags = (unsigned*)ws;  ws += (size_t)NC * 4;
    unsigned* owner = (unsigned*)ws;  ws += (size_t)NC * 4;
    int*   cell_t   = (int*)ws;       ws += (size_t)nT * 4;
    float* ta_t     = (float*)ws;     ws += (size_t)nT * 4;
    float* ar_t     = (float*)ws;     ws += (size_t)nT * 4;
    float* ci_t     = (float*)ws;     ws += (size_t)nT * 4;
    int nblk  = (NC + 255) / 256;
    int noblk = (nT + 63) / 64;
    float* partials = (float*)ws;     ws += (size_t)nblk * 4 * 4;
    float* objpart  = (float*)ws;

    k_init<<<nblk, 256, 0, stream>>>(flags, owner, NC);
    k_target<<<noblk, 64, 0, stream>>>(x, tgt, nT, flags, owner,
                                       cell_t, ta_t, ar_t, ci_t);
    k_objloss<<<noblk, 64, 0, stream>>>(x, flags, owner, cell_t,
                                        ta_t, ar_t, ci_t, nT, objpart);
    k_loss<<<nblk, 256, 0, stream>>>(x, flags, partials, NC);
    k_final<<<1, 256, 0, stream>>>(partials, nblk, objpart, noblk, bs, (float*)d_out);
}